// DTWAligner_51505247814355
// MI455X (gfx1250) — compile-verified
//
#include <hip/hip_runtime.h>
#include <hip/hip_bf16.h>
#include <math.h>

// ---------------------------------------------------------------------------
// Soft-DTW aligner for MI455X (gfx1250, wave32).
//   B=4, N=M=512, D=64, gamma=0.1
// Stage 1: norms of each row of x and y               (tiny)
// Stage 2: D[i,j] = sqrt(max(0, nx+ny-2*x.y)) via fp32 WMMA (16x16x4_f32),
//          stored in COMPACT ANTI-DIAGONAL-MAJOR order so stage 3 reads are
//          fully coalesced.
// Stage 3: anti-diagonal wavefront soft-DTW, 1 workgroup (512 thr) per batch,
//          ONE barrier per diagonal, coalesced D reads.
// Stage 4: trivial paths (identity) + costs into d_out
//
// Workspace layout (floats):
//   [0, 4*512*512)               : D matrices (diagonal-major, compact)
//   [4*512*512, +4*512)          : nx
//   [4*512*512+4*512, +4*512)    : ny
// ---------------------------------------------------------------------------

typedef __attribute__((ext_vector_type(2)))  float    v2f;
typedef __attribute__((ext_vector_type(8)))  float    v8f;

#define BATCH 4
#define NPTS  512
#define DIM   64
#define GAMMA 0.1f
#define DIAG_HALF 131328   /* 513*512/2 : cells on diagonals k = 2..513 */

// off(k): flat offset of diagonal k (cells (i,j), i,j>=1, i+j=k) in compact
// diagonal-major storage. idx within diagonal = i - max(1, k-512).
__device__ __forceinline__ int diag_off(int k) {
    return (k <= 513) ? (((k - 1) * (k - 2)) >> 1)
                      : (DIAG_HALF + (((1537 - k) * (k - 514)) >> 1));
}

// ---------------- Stage 1: row norms ---------------------------------------
__global__ __launch_bounds__(256) void sdtw_norms(const float* __restrict__ x,
                                                  const float* __restrict__ y,
                                                  float* __restrict__ nx,
                                                  float* __restrict__ ny) {
    int r = blockIdx.x * blockDim.x + threadIdx.x;      // 0 .. 4095
    if (r >= 2 * BATCH * NPTS) return;
    const float* src = (r < BATCH * NPTS) ? x : y;
    float*       dst = (r < BATCH * NPTS) ? nx : ny;
    int rr = (r < BATCH * NPTS) ? r : r - BATCH * NPTS;
    const float4* p = (const float4*)(src + (size_t)rr * DIM);
    float s = 0.f;
#pragma unroll
    for (int q = 0; q < DIM / 4; ++q) {
        float4 v = p[q];
        s += v.x * v.x + v.y * v.y + v.z * v.z + v.w * v.w;
    }
    dst[rr] = s;
}

// ---------------- Stage 2: distance matrix via fp32 WMMA -------------------
// 256 threads = 8 waves; each wave computes one 16x16 tile of G = X * Y^T,
// chaining 16 K=4 fp32 WMMAs, then fuses norms + sqrt and scatters D into
// the compact diagonal-major layout (throughput phase; scatter is cheap here).
__global__ __launch_bounds__(256) void sdtw_dist(const float* __restrict__ X,
                                                 const float* __restrict__ Y,
                                                 const float* __restrict__ nx,
                                                 const float* __restrict__ ny,
                                                 float* __restrict__ Dd) {
    const int wave = threadIdx.x >> 5;
    const int lane = threadIdx.x & 31;
    const int gt   = blockIdx.x * 8 + wave;     // global tile id, 0..4095
    const int b    = gt >> 10;                  // batch
    const int t    = gt & 1023;                 // tile within batch (32x32 tiles)
    const int m0   = (t >> 5) << 4;
    const int n0   = (t & 31) << 4;

    const float* Xb = X + (size_t)b * NPTS * DIM;
    const float* Yb = Y + (size_t)b * NPTS * DIM;

    const int mrow = m0 + (lane & 15);          // A: lane -> output row
    const int nrow = n0 + (lane & 15);          // B: lane -> output col (row of Y)
    const int kh   = lane >> 4;                 // which K pair this lane covers

    v8f acc = {};

    // fp32 WMMA fragments: per lane, A = float2 {X[m][ko+2*kh], X[m][ko+2*kh+1]},
    // B mirrors it with lane = output column (row of Y). Pure row-major loads.
    {
        const float* ap = Xb + (size_t)mrow * DIM + 2 * kh;
        const float* bp = Yb + (size_t)nrow * DIM + 2 * kh;
#pragma unroll
        for (int ko = 0; ko < DIM; ko += 4) {
            v2f a, bb;
            a.x  = ap[ko]; a.y  = ap[ko + 1];
            bb.x = bp[ko]; bb.y = bp[ko + 1];
            acc = __builtin_amdgcn_wmma_f32_16x16x4_f32(
                false, a, false, bb, (short)0, acc, false, false);
        }
    }

    // Epilogue: dist(i,j) = sqrt(max(0, nx[i] + ny[j] - 2*G[i,j])), scattered
    // into diagonal-major storage: Dd[b][off(i+j) + (i - lo(i+j))].
    // NOTE: storage indexes diagonals by matrix cells (i,j) starting at 1,
    // i.e. cell (row r, col c) zero-based maps to i=r+1, j=c+1, k=i+j.
    const int jcol = n0 + (lane & 15);          // zero-based column
    const float nyj = ny[b * NPTS + jcol];
    float* Db = Dd + (size_t)b * NPTS * NPTS;
#pragma unroll
    for (int v = 0; v < 8; ++v) {
        const int irow = m0 + v + 8 * kh;       // zero-based row
        const float g  = acc[v];
        const float dd = nx[b * NPTS + irow] + nyj - 2.f * g;
        const float dist = sqrtf(fmaxf(dd, 0.f));
        const int i = irow + 1, j = jcol + 1;
        const int k = i + j;
        const int lo = (k - NPTS > 1) ? (k - NPTS) : 1;
        Db[diag_off(k) + (i - lo)] = dist;
    }
}

// ---------------- Stage 3: soft-DTW wavefront ------------------------------
// One workgroup (512 threads, 16 waves) per batch. Three rotating
// anti-diagonal LDS buffers; ONE barrier per diagonal (edge threads maintain
// the INF border); D reads are dense/coalesced thanks to diagonal-major D.
__global__ __launch_bounds__(512) void sdtw_dp(const float* __restrict__ Dd,
                                               float* __restrict__ cost_out) {
    const int b = blockIdx.x;
    const float* Db = Dd + (size_t)b * NPTS * NPTS;
    __shared__ float buf0[NPTS + 1], buf1[NPTS + 1], buf2[NPTS + 1];

    const int t = threadIdx.x;
    const float INF  = __builtin_inff();
    const float invg = 1.f / GAMMA;

    // k=0 diagonal (buf0): R[0][0]=0, rest inf. k=1 (buf1): inf. buf2: inf.
    buf0[t] = INF; buf1[t] = INF; buf2[t] = INF;
    if (t == 0) {
        buf0[NPTS] = INF; buf1[NPTS] = INF; buf2[NPTS] = INF;
        buf0[0] = 0.f;
    }
    __syncthreads();

    float* cur = buf2;   // diagonal k
    float* d1  = buf1;   // diagonal k-1
    float* d2  = buf0;   // diagonal k-2
    int off = 0;         // compact offset of diagonal k in Db

    for (int k = 2; k <= 2 * NPTS; ++k) {
        const int lo = (k - NPTS > 1) ? (k - NPTS) : 1;
        const int hi = (k - 1 < NPTS) ? (k - 1) : NPTS;
        const int i  = lo + t;
        if (i <= hi) {
            const float d    = Db[off + t];          // coalesced
            const float diag = d2[i - 1];
            const float up   = d1[i - 1];
            const float left = d1[i];
            const float m = fminf(diag, fminf(up, left));
            float r = INF;
            if (m < 3.0e38f) {
                const float s = __expf((m - diag) * invg) +
                                __expf((m - up)   * invg) +
                                __expf((m - left) * invg);
                r = d + m - GAMMA * __logf(s);
            }
            cur[i] = r;
            // Maintain the INF border: only the two edge cells of this
            // diagonal can be read by later iterations without being
            // rewritten above.
            if (i == lo) cur[lo - 1] = INF;
            if (i == hi && hi < NPTS) cur[hi + 1] = INF;
        }
        __syncthreads();
        off += hi - lo + 1;
        float* tmp = d2; d2 = d1; d1 = cur; cur = tmp;   // rotate
    }

    // After the final rotation, diagonal k=1024 lives in d1; R[N][M] at i=N.
    if (t == 0) cost_out[b] = d1[NPTS];
}

// ---------------- Stage 4: paths (identity since N == M) -------------------
__global__ __launch_bounds__(256) void sdtw_paths(int* __restrict__ pout) {
    int idx = blockIdx.x * blockDim.x + threadIdx.x;   // 0..4095 covers i+j
    if (idx < 2 * BATCH * NPTS) pout[idx] = idx & (NPTS - 1);
}

// ---------------------------------------------------------------------------
extern "C" void kernel_launch(void* const* d_in, const int* in_sizes, int n_in,
                              void* d_out, int out_size, void* d_ws, size_t ws_size,
                              hipStream_t stream) {
    const float* x = (const float*)d_in[0];   // (4, 512, 64) f32
    const float* y = (const float*)d_in[1];   // (4, 512, 64) f32

    float* ws  = (float*)d_ws;
    float* Dd  = ws;                                        // 4*512*512 floats
    float* nx  = ws + (size_t)BATCH * NPTS * NPTS;          // 4*512
    float* ny  = nx + BATCH * NPTS;                         // 4*512

    int*   path_out = (int*)d_out;                          // [0, 4096): i then j
    float* cost_out = (float*)d_out + 2 * BATCH * NPTS;     // [4096, 4100)

    // Stage 1: row norms (4096 rows)
    sdtw_norms<<<(2 * BATCH * NPTS + 255) / 256, 256, 0, stream>>>(x, y, nx, ny);

    // Stage 2: 4096 16x16 tiles, 8 waves per block -> 512 blocks
    sdtw_dist<<<(BATCH * 1024) / 8, 256, 0, stream>>>(x, y, nx, ny, Dd);

    // Stage 4 (independent of stage 3): paths
    sdtw_paths<<<(2 * BATCH * NPTS + 255) / 256, 256, 0, stream>>>(path_out);

    // Stage 3: one workgroup per batch
    sdtw_dp<<<BATCH, NPTS, 0, stream>>>(Dd, cost_out);
}